// CanadarmJacob_73839077753193
// MI455X (gfx1250) — compile-verified
//
#include <hip/hip_runtime.h>
#include <stdint.h>
#include <utility>

#define NS 2048
#define NH 128
#define NPROB (NS * NH)     // 262144 problems
#define TPB 256             // 8 wave32 per block, one problem per thread

// ---------------------------------------------------------------------------
// compile-time unroll helper (so asm "n" offsets are integer-constant exprs)
// ---------------------------------------------------------------------------
template <class F, int... I>
__device__ __forceinline__ void sfor_impl(F&& f, std::integer_sequence<int, I...>) {
    (f(std::integral_constant<int, I>{}), ...);
}
template <int N, class F>
__device__ __forceinline__ void sfor(F&& f) {
    sfor_impl(static_cast<F&&>(f), std::make_integer_sequence<int, N>{});
}

// ---------------------------------------------------------------------------
// gfx1250 async global<->LDS ops (ASYNCcnt path, CDNA5 ISA)
// SADDR form: global addr = SGPR64 + voffset + INST_OFFSET
//             lds addr    = VDST   + INST_OFFSET          (offset added to both)
// ---------------------------------------------------------------------------
__device__ __forceinline__ unsigned lds_addr_u32(const void* p) {
    return (unsigned)(unsigned long long)(uintptr_t)p;
}
template <int OFF>
__device__ __forceinline__ void async_load_b32_s(unsigned lds, unsigned voff, const void* sbase) {
    asm volatile("global_load_async_to_lds_b32 %0, %1, %2 offset:%3"
                 :: "v"(lds), "v"(voff), "s"(sbase), "n"(OFF) : "memory");
}
template <int OFF>
__device__ __forceinline__ void async_load_b128_s(unsigned lds, unsigned voff, const void* sbase) {
    asm volatile("global_load_async_to_lds_b128 %0, %1, %2 offset:%3"
                 :: "v"(lds), "v"(voff), "s"(sbase), "n"(OFF) : "memory");
}
template <int OFF>
__device__ __forceinline__ void async_store_b128_s(unsigned voff, unsigned lds, const void* sbase) {
    asm volatile("global_store_async_from_lds_b128 %0, %1, %2 offset:%3"
                 :: "v"(voff), "v"(lds), "s"(sbase), "n"(OFF) : "memory");
}
__device__ __forceinline__ void wait_async0() {
    asm volatile("s_wait_asynccnt 0x0" ::: "memory");
}

__global__ __launch_bounds__(TPB)
void canadarm_jacob_kernel(const float* __restrict__ com_list,     // (b,h,3,7)
                           const float* __restrict__ link_pose,    // (b,h,4,4,9)
                           const float* __restrict__ jacobian,     // (b,h,6,7)
                           float* __restrict__ out)                // (b,h,6,7)
{
    // LDS layout (floats):
    //   [0     .. 5376 )  com tile       (256 problems * 21 words)
    //   [5376  .. 10752)  jac tile       (256 * 21 words, rows 0..2 only)
    //   [10752 .. 16128)  link-pose tile (256 * 21 words, gathered [r,3,i])
    // After compute, [0..10752) is reused to stage the 256*42-word output.
    __shared__ __align__(16) float smem[16128];   // 64512 B -> ~5 blocks / 320KB WGP

    const int      t      = threadIdx.x;
    const int      block0 = blockIdx.x * TPB;     // first problem of block
    const unsigned sbase  = lds_addr_u32(smem);
    const unsigned v16    = (unsigned)t * 16u;    // vec4 lane offset for bulk copies

    // ---- stage com: block span = 256*84 B contiguous, 16B aligned -> b128 bulk.
    //      1344 vec4 = 5 full rounds of 256 + 64 tail. saddr uniform, offsets imm.
    {
        const float* g = com_list + (size_t)block0 * 21;
        sfor<5>([&](auto IT) {
            constexpr int off = decltype(IT)::value * 4096;
            async_load_b128_s<off>(sbase + v16, v16, g);
        });
        if (t < 64) async_load_b128_s<20480>(sbase + v16, v16, g);
    }
    // ---- gather jacobian rows 0..2 (first 21 of 42 words per problem).
    //      global delta per word == LDS delta per word == 4*k -> one imm per k.
    {
        const float*   g    = jacobian + (size_t)block0 * 42;
        const unsigned voff = (unsigned)t * 168u;
        const unsigned lj   = sbase + (5376u + (unsigned)t * 21u) * 4u;
        sfor<21>([&](auto K) {
            constexpr int off = decltype(K)::value * 4;
            async_load_b32_s<off>(lj, voff, g);
        });
    }
    // ---- gather link_pose[.., r, 3, i]: global word 36r+27+i, LDS word r*7+i.
    //      Fold the stride difference into 3 uniform SGPR bases:
    //      base_r = lp + block0*144 + (29r+27); then imm = 4*(r*7+i) fits both.
    {
        const float*   g    = link_pose + (size_t)block0 * 144;
        const unsigned voff = (unsigned)t * 576u;
        const unsigned ll   = sbase + (10752u + (unsigned)t * 21u) * 4u;
        sfor<7>([&](auto I) {
            constexpr int off = decltype(I)::value * 4;           // r = 0
            async_load_b32_s<off>(ll, voff, g + 27);
        });
        sfor<7>([&](auto I) {
            constexpr int off = (7 + decltype(I)::value) * 4;     // r = 1
            async_load_b32_s<off>(ll, voff, g + 56);              // 29*1+27
        });
        sfor<7>([&](auto I) {
            constexpr int off = (14 + decltype(I)::value) * 4;    // r = 2
            async_load_b32_s<off>(ll, voff, g + 85);              // 29*2+27
        });
    }
    wait_async0();
    __syncthreads();

    // ------------------------- constants -------------------------
    constexpr float MASS[7] = {105.98f, 105.98f, 314.98f, 279.2f, 105.98f, 105.98f, 243.66f};
    constexpr float SUMM = 105.98f + 105.98f + 314.98f + 279.2f + 105.98f + 105.98f + 243.66f;
    constexpr float TM   = SUMM + 100000.0f + 243.66f;       // total system mass
    constexpr float BCZ  = 6.65f * (243.66f / (100000.0f + 243.66f));
    const float DIAG[7][3] = {
        {12.19f, 12.19f, 3.061f}, {12.19f, 12.19f, 3.061f}, {15.41f, 2094.71f, 2103.19f},
        {9.522f, 1966.28f, 1966.28f}, {8.305f, 3.061f, 8.0386f}, {12.13f, 12.13f, 3.061f},
        {9.336f, 44.41f, 44.41f}};
    float DS[7][3];   // suffix sums of inertia diagonals (constant-folded)
    {
        float a0 = 0.f, a1 = 0.f, a2 = 0.f;
        #pragma unroll
        for (int j = 6; j >= 0; --j) {
            a0 += DIAG[j][0]; a1 += DIAG[j][1]; a2 += DIAG[j][2];
            DS[j][0] = a0; DS[j][1] = a1; DS[j][2] = a2;
        }
    }
    const float SD0 = DS[0][0] + 69585.02f;     // sum(inertial)+I0 diagonal
    const float SD1 = DS[0][1] + 69585.02f;
    const float SD2 = DS[0][2] + 66666.664f;

    // ------------------ per-problem data from LDS ------------------
    const float* C  = smem + t * 21;            // com[c*7+i]
    const float* Jm = smem + 5376 + t * 21;     // J[c*7+j]
    const float* Lp = smem + 10752 + t * 21;    // link pos [r*7+i]

    float com[3][7], rpt[7][3], roi[7][3], Jt[7][3];
    #pragma unroll
    for (int c = 0; c < 3; ++c) {
        #pragma unroll
        for (int i = 0; i < 7; ++i) com[c][i] = C[c * 7 + i];
    }
    #pragma unroll
    for (int i = 0; i < 7; ++i) {
        rpt[i][0] = com[0][i] - Lp[i];
        rpt[i][1] = com[1][i] - Lp[7 + i];
        rpt[i][2] = com[2][i] - Lp[14 + i];
        roi[i][0] = com[0][i];
        roi[i][1] = com[1][i];
        roi[i][2] = com[2][i] - BCZ;
        Jt[i][0]  = Jm[i];
        Jt[i][1]  = Jm[7 + i];
        Jt[i][2]  = Jm[14 + i];
    }

    // system COM -> r_og
    float rog[3];
    #pragma unroll
    for (int c = 0; c < 3; ++c) {
        float s = 0.f;
        #pragma unroll
        for (int i = 0; i < 7; ++i) s += com[c][i] * MASS[i];
        rog[c] = s / TM;
    }
    rog[2] -= BCZ;

    // ---- suffix-sum sweep j = 6..0 (factorized J_tw / Hwp1 / Hwp2 / H_theta) ----
    float W0 = 0.f, W1 = 0.f, W2 = 0.f, As = 0.f;
    float M00 = 0.f, M01 = 0.f, M02 = 0.f, M10 = 0.f, M11 = 0.f, M12 = 0.f,
          M20 = 0.f, M21 = 0.f, M22 = 0.f;   // M[a][b] = sum m * rpt[a]*roi[b]
    float jtw[7][3], hth[7][3];
    #pragma unroll
    for (int j = 6; j >= 0; --j) {
        const float m  = MASS[j];
        const float rx = rpt[j][0], ry = rpt[j][1], rz = rpt[j][2];
        const float ox = roi[j][0], oy = roi[j][1], oz = roi[j][2];
        W0 += m * rx; W1 += m * ry; W2 += m * rz;
        As += m * (ox * rx + oy * ry + oz * rz);
        M00 += m * rx * ox; M01 += m * rx * oy; M02 += m * rx * oz;
        M10 += m * ry * ox; M11 += m * ry * oy; M12 += m * ry * oz;
        M20 += m * rz * ox; M21 += m * rz * oy; M22 += m * rz * oz;

        const float jx = Jt[j][0], jy = Jt[j][1], jz = Jt[j][2];
        // J_tw[:,j] = Jt_j x W_j
        const float t0 = jy * W2 - jz * W1;
        const float t1 = jz * W0 - jx * W2;
        const float t2 = jx * W1 - jy * W0;
        jtw[j][0] = t0; jtw[j][1] = t1; jtw[j][2] = t2;
        // Hwp1 + Hwp2 = DS_j (.) Jt + A_j*Jt - M_j*Jt
        const float h0 = jx * DS[j][0] + As * jx - (M00 * jx + M01 * jy + M02 * jz);
        const float h1 = jy * DS[j][1] + As * jy - (M10 * jx + M11 * jy + M12 * jz);
        const float h2 = jz * DS[j][2] + As * jz - (M20 * jx + M21 * jy + M22 * jz);
        // H_theta[:,j] = Hwp1+Hwp2 - rog x J_tw[:,j]
        hth[j][0] = h0 - (rog[1] * t2 - rog[2] * t1);
        hth[j][1] = h1 - (rog[2] * t0 - rog[0] * t2);
        hth[j][2] = h2 - (rog[0] * t1 - rog[1] * t0);
    }

    // ---- H_s (symmetric) and its analytic inverse ----
    const float K  = SUMM - TM;   // = -(100000+243.66)
    const float r2 = rog[0] * rog[0] + rog[1] * rog[1] + rog[2] * rog[2];
    const float H00 = SD0 + K * (r2 - rog[0] * rog[0]);
    const float H11 = SD1 + K * (r2 - rog[1] * rog[1]);
    const float H22 = SD2 + K * (r2 - rog[2] * rog[2]);
    const float H01 = -K * rog[0] * rog[1];
    const float H02 = -K * rog[0] * rog[2];
    const float H12 = -K * rog[1] * rog[2];
    const float A_  = H11 * H22 - H12 * H12;
    const float B_  = H02 * H12 - H01 * H22;
    const float C_  = H01 * H12 - H02 * H11;
    const float det = H00 * A_ + H01 * B_ + H02 * C_;
    const float id  = 1.0f / det;
    const float i00 = A_ * id, i01 = B_ * id, i02 = C_ * id;
    const float i11 = (H00 * H22 - H02 * H02) * id;
    const float i12 = (H02 * H01 - H00 * H12) * id;
    const float i22 = (H00 * H11 - H01 * H01) * id;

    // ---- stage output into LDS (reuse com+jac tiles), then async write-back ----
    __syncthreads();                      // all LDS reads of input tiles done
    float*          O   = smem + t * 42;  // out[r*7+j] for this problem
    constexpr float ITM = 1.0f / TM;
    #pragma unroll
    for (int j = 0; j < 7; ++j) {
        const float v0 = i00 * hth[j][0] + i01 * hth[j][1] + i02 * hth[j][2];
        const float v1 = i01 * hth[j][0] + i11 * hth[j][1] + i12 * hth[j][2];
        const float v2 = i02 * hth[j][0] + i12 * hth[j][1] + i22 * hth[j][2];
        O[0 * 7 + j] = -(jtw[j][0] * ITM + (rog[1] * v2 - rog[2] * v1));
        O[1 * 7 + j] = -(jtw[j][1] * ITM + (rog[2] * v0 - rog[0] * v2));
        O[2 * 7 + j] = -(jtw[j][2] * ITM + (rog[0] * v1 - rog[1] * v0));
        O[3 * 7 + j] = -v0;
        O[4 * 7 + j] = -v1;
        O[5 * 7 + j] = -v2;
    }
    __syncthreads();
    // async store-back: 2688 vec4 = 10 full rounds + 128 tail, contiguous, coalesced
    {
        const float* g = out + (size_t)block0 * 42;
        sfor<10>([&](auto IT) {
            constexpr int off = decltype(IT)::value * 4096;
            async_store_b128_s<off>(v16, sbase + v16, g);
        });
        if (t < 128) async_store_b128_s<40960>(v16, sbase + v16, g);
    }
    wait_async0();
}

extern "C" void kernel_launch(void* const* d_in, const int* in_sizes, int n_in,
                              void* d_out, int out_size, void* d_ws, size_t ws_size,
                              hipStream_t stream) {
    const float* com_list  = (const float*)d_in[0];   // (2048,128,3,7)
    const float* link_pose = (const float*)d_in[1];   // (2048,128,4,4,9)
    const float* jacobian  = (const float*)d_in[2];   // (2048,128,6,7)
    float*       out       = (float*)d_out;           // (2048,128,6,7)

    dim3 grid(NPROB / TPB);   // 1024 blocks
    dim3 block(TPB);
    canadarm_jacob_kernel<<<grid, block, 0, stream>>>(com_list, link_pose, jacobian, out);
}